// GraphAttentionLayer_57277683859483
// MI455X (gfx1250) — compile-verified
//
#include <hip/hip_runtime.h>

typedef __attribute__((ext_vector_type(2))) float v2f;
typedef __attribute__((ext_vector_type(8))) float v8f;

#define NN 8192
#define DIN 256
#define DOUT 128
#define SLOPE 0.2f
#define EPSN 1e-5f

__device__ __forceinline__ float lrelu(float v) { return v > 0.f ? v : SLOPE * v; }

// ---------------------------------------------------------------------------
// Kernel 1: proj[h] = X @ W[h]  (h=0,1)  and  resid = X @ W_res^T
// fp32 WMMA 16x16x4. Each wave owns one 16x16 output tile.
// grid = (512 m-tiles, 3), block = 256 (8 waves); wave -> f-tile t = by*8+w, t in [0,24)
// B source selection hoisted to a uniform (base,stride) pair -> branch-free K loop.
// ---------------------------------------------------------------------------
__global__ __launch_bounds__(256) void k_proj(const float* __restrict__ x,
                                              const float* __restrict__ W,
                                              const float* __restrict__ Wres,
                                              float* __restrict__ proj,
                                              float* __restrict__ resid) {
  const int lane = threadIdx.x & 31;
  const int wv   = threadIdx.x >> 5;
  const int t    = blockIdx.y * 8 + wv;           // 0..23
  const int m0   = blockIdx.x * 16;
  const int hi   = lane >> 4;                     // 0: K0/K1 pair, 1: K2/K3 pair
  const int l15  = lane & 15;
  const int o0   = hi ? 2 : 0;                    // b.x K offset
  const int o1   = hi ? 3 : 1;                    // b.y K offset

  const float* arow = x + (m0 + l15) * DIN + 2 * hi;  // contiguous (Klo,Khi) pair

  const float* bbase;
  int bstride;
  float* dst;
  int fc;
  if (t < 16) {
    fc      = (t & 7) * 16 + l15;
    bbase   = W + (size_t)(t >> 3) * DIN * DOUT + fc;
    bstride = DOUT;
    dst     = proj + (size_t)(t >> 3) * NN * DOUT;
  } else {
    fc      = (t - 16) * 16 + l15;
    bbase   = Wres + (size_t)fc * DIN;
    bstride = 1;
    dst     = resid;
  }

  v8f c = {};
  #pragma unroll 4
  for (int kb = 0; kb < DIN; kb += 4) {
    v2f a = *(const v2f*)(arow + kb);             // global_load_b64
    v2f b;
    b.x = bbase[(kb + o0) * bstride];
    b.y = bbase[(kb + o1) * bstride];
    c = __builtin_amdgcn_wmma_f32_16x16x4_f32(false, a, false, b, (short)0, c, false, false);
  }
  #pragma unroll
  for (int r = 0; r < 8; ++r) {
    const int m = m0 + r + (hi ? 8 : 0);
    dst[m * DOUT + fc] = c[r];
  }
}

// ---------------------------------------------------------------------------
// Kernel 2: s_src[h,n] = proj[h,n,:].a_src[h]   s_tgt likewise
// ---------------------------------------------------------------------------
__global__ __launch_bounds__(256) void k_scores(const float* __restrict__ proj,
                                                const float* __restrict__ a_src,
                                                const float* __restrict__ a_tgt,
                                                float* __restrict__ s_src,
                                                float* __restrict__ s_tgt) {
  const int idx = blockIdx.x * blockDim.x + threadIdx.x;   // 2*8192
  if (idx >= 2 * NN) return;
  const int h = idx >> 13, n = idx & (NN - 1);
  const float* p  = proj + ((size_t)h * NN + n) * DOUT;
  const float* as = a_src + h * DOUT;
  const float* at = a_tgt + h * DOUT;
  float ss = 0.f, st = 0.f;
  #pragma unroll 8
  for (int f = 0; f < DOUT; ++f) { const float v = p[f]; ss += v * as[f]; st += v * at[f]; }
  s_src[idx] = ss;
  s_tgt[idx] = st;
}

// ---------------------------------------------------------------------------
// Kernel 3: per-column online softmax stats (softmax over rows, axis=1),
// partial over 1024-row blocks. grid=(32,8) block=256; one column per thread.
// ---------------------------------------------------------------------------
__global__ __launch_bounds__(256) void k_colstats_part(const float* __restrict__ mask,
                                                       const float* __restrict__ s_src,
                                                       const float* __restrict__ s_tgt,
                                                       float* __restrict__ pmax,
                                                       float* __restrict__ psum) {
  __shared__ float ssl[2][1024];
  const int n  = blockIdx.x * 256 + threadIdx.x;
  const int rb = blockIdx.y;
  const int mbeg = rb * 1024;
  for (int i = threadIdx.x; i < 2048; i += 256) {
    const int h = i >> 10, r = i & 1023;
    ssl[h][r] = s_src[h * NN + mbeg + r];
  }
  __syncthreads();
  const float st0 = s_tgt[n], st1 = s_tgt[NN + n];
  float mx0 = -3.4e38f, mx1 = -3.4e38f, sm0 = 0.f, sm1 = 0.f;
  for (int r = 0; r < 1024; ++r) {
    const float mk = __builtin_nontemporal_load(&mask[(size_t)(mbeg + r) * NN + n]);
    const float z0 = mk + lrelu(ssl[0][r] + st0);
    const float z1 = mk + lrelu(ssl[1][r] + st1);
    const float a0 = fmaxf(mx0, z0);
    sm0 = sm0 * __expf(mx0 - a0) + __expf(z0 - a0); mx0 = a0;
    const float a1 = fmaxf(mx1, z1);
    sm1 = sm1 * __expf(mx1 - a1) + __expf(z1 - a1); mx1 = a1;
  }
  pmax[(0 * 8 + rb) * NN + n] = mx0; psum[(0 * 8 + rb) * NN + n] = sm0;
  pmax[(1 * 8 + rb) * NN + n] = mx1; psum[(1 * 8 + rb) * NN + n] = sm1;
}

__global__ __launch_bounds__(256) void k_colstats_merge(const float* __restrict__ pmax,
                                                        const float* __restrict__ psum,
                                                        float* __restrict__ colmax,
                                                        float* __restrict__ colsum) {
  const int idx = blockIdx.x * blockDim.x + threadIdx.x;   // 2*8192
  if (idx >= 2 * NN) return;
  const int h = idx >> 13, n = idx & (NN - 1);
  float mx = -3.4e38f;
  #pragma unroll
  for (int rb = 0; rb < 8; ++rb) mx = fmaxf(mx, pmax[(h * 8 + rb) * NN + n]);
  float sm = 0.f;
  #pragma unroll
  for (int rb = 0; rb < 8; ++rb)
    sm += psum[(h * 8 + rb) * NN + n] * __expf(pmax[(h * 8 + rb) * NN + n] - mx);
  colmax[idx] = mx;
  colsum[idx] = sm;
}

// ---------------------------------------------------------------------------
// Kernel 4: vals[m,f] = 0.5 * sum_h sum_n attn[h,m,n] * proj[h,n,f]
// attn recomputed on the fly from streamed mask + rank-1 scores + col stats.
// grid = 256 (32 rows each), block = 256 (8 waves); wave -> (mt = w&1, fb=(w>>1)*32),
// 2 f-tiles per wave. K-loop over n in chunks of 32; weights staged in LDS
// (row pad 33, coprime with 64 banks -> conflict-free A-fragment reads).
// ---------------------------------------------------------------------------
#define NCH 32
__global__ __launch_bounds__(256) void k_attn_apply(const float* __restrict__ mask,
                                                    const float* __restrict__ proj,
                                                    const float* __restrict__ s_src,
                                                    const float* __restrict__ s_tgt,
                                                    const float* __restrict__ colmax,
                                                    const float* __restrict__ colsum,
                                                    float* __restrict__ vals) {
  __shared__ float wt[2][32][NCH + 1];
  __shared__ float ss[2][32];
  __shared__ float cmx[2][NCH], crs[2][NCH], stl[2][NCH];
  const int tid  = threadIdx.x;
  const int lane = tid & 31, wv = tid >> 5;
  const int hi   = lane >> 4, l15 = lane & 15;
  const int o0   = hi ? 2 : 0;
  const int o1   = hi ? 3 : 1;
  const int mb   = blockIdx.x * 32;
  const int mt   = wv & 1;
  const int fb   = (wv >> 1) * 32;
  if (tid < 64) { const int h = tid >> 5, r = tid & 31; ss[h][r] = s_src[h * NN + mb + r]; }
  v8f acc0 = {}, acc1 = {};
  for (int nc = 0; nc < NN; nc += NCH) {
    __syncthreads();   // protects wt/stl reuse from previous iteration
    if (tid < 64) {
      const int h = tid >> 5, j = tid & 31;
      cmx[h][j] = colmax[h * NN + nc + j];
      crs[h][j] = 1.0f / colsum[h * NN + nc + j];
      stl[h][j] = s_tgt[h * NN + nc + j];
    }
    __syncthreads();
    #pragma unroll
    for (int h = 0; h < 2; ++h) {
      #pragma unroll
      for (int i = 0; i < 4; ++i) {
        const int e = i * 256 + tid;           // 0..1023  (32m x 32n per head)
        const int m = e >> 5, j = e & 31;
        const float mk = __builtin_nontemporal_load(&mask[(size_t)(mb + m) * NN + nc + j]);
        const float z  = mk + lrelu(ss[h][m] + stl[h][j]);
        wt[h][m][j] = __expf(z - cmx[h][j]) * crs[h][j];
      }
    }
    __syncthreads();
    #pragma unroll
    for (int h = 0; h < 2; ++h) {
      const float* ph = proj + (size_t)h * NN * DOUT;
      #pragma unroll
      for (int ks = 0; ks < 8; ++ks) {         // 8 K-steps of 4 within the 32-chunk
        v2f a;
        a.x = wt[h][mt * 16 + l15][ks * 4 + o0];
        a.y = wt[h][mt * 16 + l15][ks * 4 + o1];
        {
          v2f b;
          const int f = fb + l15;
          b.x = ph[(size_t)(nc + ks * 4 + o0) * DOUT + f];
          b.y = ph[(size_t)(nc + ks * 4 + o1) * DOUT + f];
          acc0 = __builtin_amdgcn_wmma_f32_16x16x4_f32(false, a, false, b, (short)0, acc0, false, false);
        }
        {
          v2f b;
          const int f = fb + 16 + l15;
          b.x = ph[(size_t)(nc + ks * 4 + o0) * DOUT + f];
          b.y = ph[(size_t)(nc + ks * 4 + o1) * DOUT + f];
          acc1 = __builtin_amdgcn_wmma_f32_16x16x4_f32(false, a, false, b, (short)0, acc1, false, false);
        }
      }
    }
  }
  #pragma unroll
  for (int r = 0; r < 8; ++r) {
    const int m = mb + mt * 16 + r + (hi ? 8 : 0);
    vals[m * DOUT + fb + l15]      = 0.5f * acc0[r];
    vals[m * DOUT + fb + 16 + l15] = 0.5f * acc1[r];
  }
}

// ---------------------------------------------------------------------------
// Kernels 5a/5b: deterministic two-stage sum / sum-of-squares reduction
// ---------------------------------------------------------------------------
__global__ __launch_bounds__(256) void k_redpart(const float* __restrict__ vals,
                                                 float* __restrict__ partials) {
  __shared__ float s1[256], s2[256];
  const int tid = threadIdx.x;
  float a = 0.f, b = 0.f;
  for (int i = blockIdx.x * 256 + tid; i < NN * DOUT; i += 256 * 256) {
    const float v = vals[i]; a += v; b += v * v;
  }
  s1[tid] = a; s2[tid] = b;
  __syncthreads();
  for (int s = 128; s > 0; s >>= 1) {
    if (tid < s) { s1[tid] += s1[tid + s]; s2[tid] += s2[tid + s]; }
    __syncthreads();
  }
  if (tid == 0) { partials[blockIdx.x] = s1[0]; partials[256 + blockIdx.x] = s2[0]; }
}

__global__ __launch_bounds__(256) void k_redfinal(const float* __restrict__ partials,
                                                  float* __restrict__ stats) {
  __shared__ float s1[256], s2[256];
  const int tid = threadIdx.x;
  s1[tid] = partials[tid]; s2[tid] = partials[256 + tid];
  __syncthreads();
  for (int s = 128; s > 0; s >>= 1) {
    if (tid < s) { s1[tid] += s1[tid + s]; s2[tid] += s2[tid + s]; }
    __syncthreads();
  }
  if (tid == 0) {
    const float inv = 1.0f / ((float)NN * (float)DOUT);
    const float mu  = s1[0] * inv;
    const float var = s2[0] * inv - mu * mu;
    stats[0] = mu;
    stats[1] = rsqrtf(var + EPSN);
  }
}

// ---------------------------------------------------------------------------
// Kernel 6: out = elu( (vals - mu)*rnorm + resid )
// ---------------------------------------------------------------------------
__global__ __launch_bounds__(256) void k_final(const float* __restrict__ vals,
                                               const float* __restrict__ resid,
                                               const float* __restrict__ stats,
                                               float* __restrict__ out) {
  const int i = blockIdx.x * blockDim.x + threadIdx.x;
  if (i >= NN * DOUT) return;
  const float mu = stats[0], rn = stats[1];
  const float v = (vals[i] - mu) * rn + resid[i];
  out[i] = v > 0.f ? v : (__expf(v) - 1.0f);
}

// ---------------------------------------------------------------------------
// Workspace layout (floats):
//   proj  [2][8192][128] @ 0         (2097152)
//   resid [8192][128]    @ 2097152   (1048576)
//   vals  [8192][128]    @ 3145728   (1048576)
//   s_src [2][8192]      @ 4194304
//   s_tgt [2][8192]      @ 4210688
//   pmax  [2][8][8192]   @ 4227072
//   psum  [2][8][8192]   @ 4358144
//   colmax[2][8192]      @ 4489216
//   colsum[2][8192]      @ 4505600
//   partials [512]       @ 4521984
//   stats [2]            @ 4522496     total ~18.1 MB
// ---------------------------------------------------------------------------
extern "C" void kernel_launch(void* const* d_in, const int* in_sizes, int n_in,
                              void* d_out, int out_size, void* d_ws, size_t ws_size,
                              hipStream_t stream) {
  const float* x     = (const float*)d_in[0];
  const float* mask  = (const float*)d_in[1];
  const float* W     = (const float*)d_in[2];
  const float* a_src = (const float*)d_in[3];
  const float* a_tgt = (const float*)d_in[4];
  const float* Wres  = (const float*)d_in[5];
  float* out = (float*)d_out;
  float* ws  = (float*)d_ws;

  float* proj     = ws;
  float* resid    = ws + 2097152;
  float* vals     = ws + 3145728;
  float* ssrc     = ws + 4194304;
  float* stgt     = ws + 4210688;
  float* pmax     = ws + 4227072;
  float* psum     = ws + 4358144;
  float* cmax     = ws + 4489216;
  float* csum     = ws + 4505600;
  float* partials = ws + 4521984;
  float* stats    = ws + 4522496;

  k_proj<<<dim3(512, 3), 256, 0, stream>>>(x, W, Wres, proj, resid);
  k_scores<<<64, 256, 0, stream>>>(proj, a_src, a_tgt, ssrc, stgt);
  k_colstats_part<<<dim3(32, 8), 256, 0, stream>>>(mask, ssrc, stgt, pmax, psum);
  k_colstats_merge<<<64, 256, 0, stream>>>(pmax, psum, cmax, csum);
  k_attn_apply<<<256, 256, 0, stream>>>(mask, proj, ssrc, stgt, cmax, csum, vals);
  k_redpart<<<256, 256, 0, stream>>>(vals, partials);
  k_redfinal<<<1, 256, 0, stream>>>(partials, stats);
  k_final<<<4096, 256, 0, stream>>>(vals, resid, stats, out);

  // tuple output: (out, conn_mask) concatenated flat
  hipMemcpyAsync(out + (size_t)NN * DOUT, mask, (size_t)NN * NN * sizeof(float),
                 hipMemcpyDeviceToDevice, stream);
}